// TriangulateHead_67731634258194
// MI455X (gfx1250) — compile-verified
//
#include <hip/hip_runtime.h>

#define NJ 20
#define ROWLEN 4096

// flat float offsets in d_out: (kp_3d, res_triang, kpc, kph)
#define OFF_KP3D 0
#define OFF_RES  3840
#define OFF_KPC  5120
#define OFF_KPH  28160

typedef __attribute__((ext_vector_type(2))) float v2f;
typedef __attribute__((ext_vector_type(8))) float v8f;

// ---------------------------------------------------------------------------
// Kernel 1: per heatmap row (n,j): softmax soft-argmax. HBM-bound streaming:
// 256 threads x 16 f32 (4x GLOBAL_LOAD_B128), data kept in registers.
// ---------------------------------------------------------------------------
__global__ __launch_bounds__(256) void softargmax_kernel(const float* __restrict__ hm,
                                                         float* __restrict__ out)
{
    const int row = blockIdx.x;                 // n*NJ + j
    const float4* p = (const float4*)(hm + (size_t)row * ROWLEN);
    const int t = threadIdx.x;
    const int lane = t & 31, wave = t >> 5;

    float4 v[4];
    float m = -3.4e38f;
#pragma unroll
    for (int k = 0; k < 4; ++k) {
        float4 x = p[t + k * 256];
        x.x *= 100.f; x.y *= 100.f; x.z *= 100.f; x.w *= 100.f;
        v[k] = x;
        m = fmaxf(m, fmaxf(fmaxf(x.x, x.y), fmaxf(x.z, x.w)));
    }

    __shared__ float red[4][8];
    // wave-level max, then cross-wave via LDS
#pragma unroll
    for (int off = 16; off > 0; off >>= 1) m = fmaxf(m, __shfl_xor(m, off, 32));
    if (lane == 0) red[0][wave] = m;
    __syncthreads();
    if (t == 0) {
        float mm = red[0][0];
#pragma unroll
        for (int i = 1; i < 8; ++i) mm = fmaxf(mm, red[0][i]);
        red[0][0] = mm;
    }
    __syncthreads();
    m = red[0][0];

    float s = 0.f, sx = 0.f, sy = 0.f;
#pragma unroll
    for (int k = 0; k < 4; ++k) {
        const int idx = (t + k * 256) * 4;      // element index in [0,4096)
        const float col = (float)(idx & 63);    // W index (4 consecutive, no wrap)
        const float rw  = (float)(idx >> 6);    // H index (constant across float4)
        float e0 = __expf(v[k].x - m);
        float e1 = __expf(v[k].y - m);
        float e2 = __expf(v[k].z - m);
        float e3 = __expf(v[k].w - m);
        float es = e0 + e1 + e2 + e3;
        s  += es;
        sx += e0 * col + e1 * (col + 1.f) + e2 * (col + 2.f) + e3 * (col + 3.f);
        sy += es * rw;
    }
#pragma unroll
    for (int off = 16; off > 0; off >>= 1) {
        s  += __shfl_xor(s,  off, 32);
        sx += __shfl_xor(sx, off, 32);
        sy += __shfl_xor(sy, off, 32);
    }
    if (lane == 0) { red[1][wave] = s; red[2][wave] = sx; red[3][wave] = sy; }
    __syncthreads();
    if (t == 0) {
        float S = 0.f, SX = 0.f, SY = 0.f;
#pragma unroll
        for (int i = 0; i < 8; ++i) { S += red[1][i]; SX += red[2][i]; SY += red[3][i]; }
        const float invS = 1.f / S;
        const float x = SX * invS, y = SY * invS, conf = invS; // conf = max/sum of softmax
        float* kph = out + OFF_KPH + row * 3;
        float* kpc = out + OFF_KPC + row * 3;
        kph[0] = x;       kph[1] = y;       kph[2] = conf;
        kpc[0] = 4.f * x; kpc[1] = 4.f * y; kpc[2] = conf;    // 256/64 scale
    }
}

// ---------------------------------------------------------------------------
// Kernel 2: one wave per (b,j). Build 12x4 A, Gram = A^T A via 3x
// V_WMMA_F32_16X16X4_F32 accumulations, 4x4 Jacobi eigensolve for the
// null-space vector, then residual sum.
// ---------------------------------------------------------------------------
__device__ __forceinline__ void jrot(float G[4][4], float V[4][4], int p, int q)
{
    const float apq = G[p][q];
    const float phi = 0.5f * atan2f(2.f * apq, G[q][q] - G[p][p]);
    const float s = __sinf(phi), c = __cosf(phi);
    float gp[4], gq[4];
#pragma unroll
    for (int k = 0; k < 4; ++k) { gp[k] = G[p][k]; gq[k] = G[q][k]; }
#pragma unroll
    for (int k = 0; k < 4; ++k) { G[p][k] = c * gp[k] - s * gq[k]; G[q][k] = s * gp[k] + c * gq[k]; }
#pragma unroll
    for (int k = 0; k < 4; ++k) { gp[k] = G[k][p]; gq[k] = G[k][q]; }
#pragma unroll
    for (int k = 0; k < 4; ++k) { G[k][p] = c * gp[k] - s * gq[k]; G[k][q] = s * gp[k] + c * gq[k]; }
#pragma unroll
    for (int k = 0; k < 4; ++k) {
        const float vp = V[k][p], vq = V[k][q];
        V[k][p] = c * vp - s * vq; V[k][q] = s * vp + c * vq;
    }
}

__global__ __launch_bounds__(128) void triangulate_kernel(const float* __restrict__ proj,
                                                          const float* __restrict__ confid,
                                                          float* __restrict__ out)
{
    const int prob = blockIdx.x * 4 + (threadIdx.x >> 5);  // 1280 problems
    const int b = prob / NJ;
    const int j = prob % NJ;
    const int lane = threadIdx.x & 31;
    const float* kpc = out + OFF_KPC;
    const float* Pb  = proj + b * 72;                      // 6 cams * 12 floats

    // normalized per-camera confidence weights
    float w[6]; float csum = 0.f;
#pragma unroll
    for (int cam = 0; cam < 6; ++cam) { w[cam] = confid[(b * 6 + cam) * NJ + j]; csum += w[cam]; }
    const float cinv = 1.f / csum;
#pragma unroll
    for (int cam = 0; cam < 6; ++cam) w[cam] = w[cam] * cinv + 1e-5f;

    // WMMA operand layout (32-bit, 16x4 A / 4x16 B): lane L holds M(or N)=L%16,
    // VGPR0 = K=(L<16?0:2), VGPR1 = K+1. Identical per-lane data for A^T (as A
    // operand) and A (as B operand), so one v2f feeds both.
    const int m  = lane & 15;
    const int kb = (lane >> 4) << 1;   // 0 or 2
    const int mk = m & 3;              // clamped column for safe addressing

    v8f acc = {0.f, 0.f, 0.f, 0.f, 0.f, 0.f, 0.f, 0.f};
#pragma unroll
    for (int c = 0; c < 3; ++c) {
        v2f av;
#pragma unroll
        for (int half = 0; half < 2; ++half) {
            const int r   = 4 * c + kb + half;   // A row 0..11
            const int cam = r >> 1;
            const int i   = r & 1;
            const float ptv = kpc[((b * 6 + cam) * NJ + j) * 3 + i];
            const float r2k = Pb[cam * 12 + 8 + mk];        // row2 of P
            const float pik = Pb[cam * 12 + i * 4 + mk];
            float a = w[cam] * (r2k * ptv - pik);
            a = (m < 4) ? a : 0.f;                          // pad M/N to 16
            if (half == 0) av.x = a; else av.y = a;
        }
        // D += (A^T chunk 16x4) x (A chunk 4x16)  => top-left 4x4 = A^T A
        acc = __builtin_amdgcn_wmma_f32_16x16x4_f32(false, av, false, av,
                                                    (short)0, acc, false, false);
    }

    // gather 4x4 Gram (rows = VGPR 0..3, cols = lanes 0..3)
    float G[4][4];
#pragma unroll
    for (int r = 0; r < 4; ++r) {
        const float ar = acc[r];
#pragma unroll
        for (int cc = 0; cc < 4; ++cc) G[r][cc] = __shfl(ar, cc, 32);
    }

    // cyclic Jacobi, 5 sweeps, fully unrolled rotations (registers only)
    float V[4][4] = {{1,0,0,0},{0,1,0,0},{0,0,1,0},{0,0,0,1}};
    for (int sweep = 0; sweep < 5; ++sweep) {
        jrot(G, V, 0, 1); jrot(G, V, 0, 2); jrot(G, V, 0, 3);
        jrot(G, V, 1, 2); jrot(G, V, 1, 3); jrot(G, V, 2, 3);
    }

    // eigenvector of smallest eigenvalue (sign irrelevant for outputs)
    int idx = 0; float dmin = G[0][0];
    if (G[1][1] < dmin) { dmin = G[1][1]; idx = 1; }
    if (G[2][2] < dmin) { dmin = G[2][2]; idx = 2; }
    if (G[3][3] < dmin) { dmin = G[3][3]; idx = 3; }
    float h[4];
#pragma unroll
    for (int k = 0; k < 4; ++k)
        h[k] = (idx == 0) ? V[k][0] : (idx == 1) ? V[k][1] : (idx == 2) ? V[k][2] : V[k][3];

    // residual: sum_n |A[n,:] . h| ; lane n (<12) recomputes its A row
    const int rr  = (lane < 12) ? lane : 0;
    const int cam = rr >> 1;
    const int i   = rr & 1;
    const float ptv = kpc[((b * 6 + cam) * NJ + j) * 3 + i];
    float rd = 0.f;
#pragma unroll
    for (int k = 0; k < 4; ++k) {
        const float a = w[cam] * (Pb[cam * 12 + 8 + k] * ptv - Pb[cam * 12 + i * 4 + k]);
        rd += a * h[k];
    }
    float res = (lane < 12) ? fabsf(rd) : 0.f;
#pragma unroll
    for (int off = 16; off > 0; off >>= 1) res += __shfl_xor(res, off, 32);

    if (lane == 0) {
        const float inv3 = 1.f / h[3];          // homo = -v; ratio cancels sign
        float* kp3 = out + OFF_KP3D + (b * NJ + j) * 3;
        kp3[0] = h[0] * inv3; kp3[1] = h[1] * inv3; kp3[2] = h[2] * inv3;
        out[OFF_RES + b * NJ + j] = res;
    }
}

extern "C" void kernel_launch(void* const* d_in, const int* in_sizes, int n_in,
                              void* d_out, int out_size, void* d_ws, size_t ws_size,
                              hipStream_t stream)
{
    const float* heatmap = (const float*)d_in[0];   // (384,20,64,64)
    const float* proj    = (const float*)d_in[1];   // (64,6,3,4)
    const float* confid  = (const float*)d_in[2];   // (384,20)
    float* out = (float*)d_out;                     // 51200 floats

    softargmax_kernel<<<7680, 256, 0, stream>>>(heatmap, out);
    triangulate_kernel<<<320, 128, 0, stream>>>(proj, confid, out);
}